// GoalFeature_83975200571439
// MI455X (gfx1250) — compile-verified
//
#include <hip/hip_runtime.h>
#include <hip/hip_bf16.h>
#include <math.h>

typedef __attribute__((ext_vector_type(2))) float v2f;
typedef __attribute__((ext_vector_type(8))) float v8f;

struct LayerP { const float *W, *b, *gm, *bt, *mu, *var; };

static __device__ __forceinline__ v8f wmma_f32(v2f a, v2f w, v8f c) {
  // D = A(16x4) x B(4x16) + C(16x16), all f32. 8-arg form per probe:
  // (neg_a, A, neg_b, B, c_mod, C, reuse_a, reuse_b)
  return __builtin_amdgcn_wmma_f32_16x16x4_f32(false, a, false, w, (short)0, c,
                                               false, false);
}

static __device__ __forceinline__ v8f vzero8() {
  v8f z = {0.f, 0.f, 0.f, 0.f, 0.f, 0.f, 0.f, 0.f};
  return z;
}

// ---------------------------------------------------------------------------
// Farthest point sampling: one block per batch. Points cached in LDS, per-point
// min-distance kept in registers, block-wide argmax with first-index tie-break
// (matches jnp.argmax). out[b*NPOINT + 0] = 0.
// ---------------------------------------------------------------------------
template<int N, int NPOINT, int TPB>
__global__ void __launch_bounds__(TPB)
fps_kernel(const float* __restrict__ xyz, int* __restrict__ out) {
  constexpr int PPT = N / TPB;
  __shared__ float px[N], py[N], pz[N];
  __shared__ float rv[TPB];
  __shared__ int   ri[TPB];
  const int b = blockIdx.x;
  const int tid = threadIdx.x;
  float dist[PPT];
#pragma unroll
  for (int j = 0; j < PPT; ++j) {
    const int i = tid + j * TPB;
    const float* p = xyz + ((size_t)b * N + i) * 3;
    px[i] = p[0]; py[i] = p[1]; pz[i] = p[2];
    dist[j] = 1e10f;
  }
  if (tid == 0) out[b * NPOINT] = 0;
  __syncthreads();
  int last = 0;
  for (int it = 1; it < NPOINT; ++it) {
    const float lx = px[last], ly = py[last], lz = pz[last];
    float bv = -1.0f; int bi = 0;
#pragma unroll
    for (int j = 0; j < PPT; ++j) {
      const int i = tid + j * TPB;
      const float dx = px[i] - lx, dy = py[i] - ly, dz = pz[i] - lz;
      float d = dx * dx + dy * dy + dz * dz;
      d = fminf(dist[j], d);
      dist[j] = d;
      if (d > bv) { bv = d; bi = i; }   // ascending i -> first-occurrence max
    }
    rv[tid] = bv; ri[tid] = bi;
    __syncthreads();
    for (int ofs = TPB / 2; ofs > 0; ofs >>= 1) {
      if (tid < ofs) {
        const float v2 = rv[tid + ofs]; const int i2 = ri[tid + ofs];
        if (v2 > rv[tid] || (v2 == rv[tid] && i2 < ri[tid])) {
          rv[tid] = v2; ri[tid] = i2;
        }
      }
      __syncthreads();
    }
    last = ri[0];
    if (tid == 0) out[b * NPOINT + it] = last;
    __syncthreads();
  }
}

// ---------------------------------------------------------------------------
// Gather sampled centers: out[b][s][:] = xyz[b][cidx[b][s]][:]
// ---------------------------------------------------------------------------
__global__ void gather_xyz_kernel(const float* __restrict__ xyz,
                                  const int* __restrict__ cidx,
                                  float* __restrict__ out,
                                  int N, int S, int total) {
  const int t = blockIdx.x * blockDim.x + threadIdx.x;
  if (t >= total) return;
  const int b = t / S;
  const int i = cidx[t];
  const float* p = xyz + ((size_t)b * N + i) * 3;
  out[t * 3 + 0] = p[0];
  out[t * 3 + 1] = p[1];
  out[t * 3 + 2] = p[2];
}

// ---------------------------------------------------------------------------
// Ball query, thread per center. Semantics match the reference: first nsample
// in-radius indices in ascending order; remaining slots padded with first hit
// (or 0 if no hits).
// ---------------------------------------------------------------------------
__global__ void ball_query_kernel(const float* __restrict__ xyz,
                                  const float* __restrict__ ctr,
                                  int* __restrict__ gidx,
                                  int N, int S, int ns, float r2, int total) {
  const int t = blockIdx.x * blockDim.x + threadIdx.x;
  if (t >= total) return;
  const int b = t / S;
  const float cx = ctr[t * 3 + 0], cy = ctr[t * 3 + 1], cz = ctr[t * 3 + 2];
  const float* base = xyz + (size_t)b * N * 3;
  int* g = gidx + (size_t)t * ns;
  int cnt = 0, first = 0; bool found = false;
  for (int i = 0; i < N; ++i) {
    const float dx = base[i * 3 + 0] - cx;
    const float dy = base[i * 3 + 1] - cy;
    const float dz = base[i * 3 + 2] - cz;
    const float d2 = dx * dx + dy * dy + dz * dz;
    if (d2 < r2) {
      if (!found) { found = true; first = i; }
      g[cnt++] = i;
      if (cnt == ns) break;
    }
  }
  const int pad = found ? first : 0;
  for (int j = cnt; j < ns; ++j) g[j] = pad;
}

// ---------------------------------------------------------------------------
// Layer-1 input feature loader (built on the fly; avoids materializing g).
// ---------------------------------------------------------------------------
template<int STAGE, int S, int NS, int NSRC, int K0, int FEATC>
__device__ __forceinline__ float load_in(const float* xyz_src,
                                         const float* centers,
                                         const int* gidx, const float* feat,
                                         int b, int s, int j, int k) {
  if (k >= K0) return 0.f;
  if constexpr (STAGE == 3) {
    // g = concat(xyz2, f2); one group of NS points per batch, no centering.
    if (k < 3) return centers[((size_t)b * NS + j) * 3 + k];
    return feat[((size_t)b * NS + j) * FEATC + (k - 3)];
  } else {
    const int idx = gidx[((size_t)b * S + s) * NS + j];
    if (k < 3)
      return xyz_src[((size_t)b * NSRC + idx) * 3 + k]
           - centers[((size_t)b * S + s) * 3 + k];
    return feat[((size_t)b * NSRC + idx) * FEATC + (k - 3)];
  }
}

// ---------------------------------------------------------------------------
// Fused set-abstraction MLP: one wave per (b,s) group. For each 16-row tile of
// grouped points: layer1 (gathered input -> LDS bufA), layer2 (bufA -> bufB),
// layer3 (bufB -> running max in LDS pool). All GEMMs on v_wmma_f32_16x16x4.
// BN(eval)+ReLU folded into scale/shift per output channel.
// ---------------------------------------------------------------------------
template<int STAGE, int S, int NS, int NSRC, int K0, int C1, int C2, int C3,
         int FEATC>
__global__ void __launch_bounds__(32)
sa_stage_kernel(const float* __restrict__ xyz_src,
                const float* __restrict__ centers,
                const int* __restrict__ gidx,
                const float* __restrict__ feat,
                LayerP L1, LayerP L2, LayerP L3,
                float* __restrict__ fout) {
  constexpr int PAD = 4;
  constexpr int KP  = (K0 + 3) & ~3;
  constexpr int ST1 = C1 + PAD;
  constexpr int ST2 = C2 + PAD;
  __shared__ float bufA[16 * ST1];
  __shared__ float bufB[16 * ST2];
  __shared__ float pool[C3];

  const int g    = blockIdx.x;
  const int b    = g / S;
  const int s    = g % S;
  const int lane = threadIdx.x & 31;
  const int half = lane >> 4;
  const int l16  = lane & 15;

  for (int c = lane; c < C3; c += 32) pool[c] = -1e30f;

  for (int t = 0; t < NS / 16; ++t) {
    // ---- layer 1: A from gathered input, W zero-padded past K0 ----
    for (int nt = 0; nt < C1 / 16; ++nt) {
      v8f acc = vzero8();
      for (int kb = 0; kb < KP; kb += 4) {
        const int k = kb + 2 * half;
        v2f a, w;
        a.x = load_in<STAGE, S, NS, NSRC, K0, FEATC>(xyz_src, centers, gidx,
                                                     feat, b, s, t * 16 + l16, k);
        a.y = load_in<STAGE, S, NS, NSRC, K0, FEATC>(xyz_src, centers, gidx,
                                                     feat, b, s, t * 16 + l16, k + 1);
        w.x = (k     < K0) ? L1.W[(size_t)k       * C1 + nt * 16 + l16] : 0.f;
        w.y = (k + 1 < K0) ? L1.W[(size_t)(k + 1) * C1 + nt * 16 + l16] : 0.f;
        acc = wmma_f32(a, w, acc);
      }
      const int n = nt * 16 + l16;
      const float sc = L1.gm[n] * rsqrtf(L1.var[n] + 1e-5f);
      const float sh = (L1.b[n] - L1.mu[n]) * sc + L1.bt[n];
#pragma unroll
      for (int j = 0; j < 8; ++j)
        bufA[(j + 8 * half) * ST1 + n] = fmaxf(acc[j] * sc + sh, 0.f);
    }
    // ---- layer 2: bufA -> bufB ----
    for (int nt = 0; nt < C2 / 16; ++nt) {
      v8f acc = vzero8();
#pragma unroll 4
      for (int kb = 0; kb < C1; kb += 4) {
        const int k = kb + 2 * half;
        v2f a, w;
        a.x = bufA[l16 * ST1 + k];
        a.y = bufA[l16 * ST1 + k + 1];
        w.x = L2.W[(size_t)k       * C2 + nt * 16 + l16];
        w.y = L2.W[(size_t)(k + 1) * C2 + nt * 16 + l16];
        acc = wmma_f32(a, w, acc);
      }
      const int n = nt * 16 + l16;
      const float sc = L2.gm[n] * rsqrtf(L2.var[n] + 1e-5f);
      const float sh = (L2.b[n] - L2.mu[n]) * sc + L2.bt[n];
#pragma unroll
      for (int j = 0; j < 8; ++j)
        bufB[(j + 8 * half) * ST2 + n] = fmaxf(acc[j] * sc + sh, 0.f);
    }
    // ---- layer 3: bufB -> max-pool over the group's samples ----
    for (int nt = 0; nt < C3 / 16; ++nt) {
      v8f acc = vzero8();
#pragma unroll 4
      for (int kb = 0; kb < C2; kb += 4) {
        const int k = kb + 2 * half;
        v2f a, w;
        a.x = bufB[l16 * ST2 + k];
        a.y = bufB[l16 * ST2 + k + 1];
        w.x = L3.W[(size_t)k       * C3 + nt * 16 + l16];
        w.y = L3.W[(size_t)(k + 1) * C3 + nt * 16 + l16];
        acc = wmma_f32(a, w, acc);
      }
      const int n = nt * 16 + l16;
      const float sc = L3.gm[n] * rsqrtf(L3.var[n] + 1e-5f);
      const float sh = (L3.b[n] - L3.mu[n]) * sc + L3.bt[n];
      float m = -1e30f;
#pragma unroll
      for (int j = 0; j < 8; ++j)
        m = fmaxf(m, fmaxf(acc[j] * sc + sh, 0.f));
      m = fmaxf(m, __shfl_xor(m, 16));       // combine row halves
      pool[n] = fmaxf(pool[n], m);           // both halves write identical value
    }
  }
  for (int c = lane; c < C3; c += 32) fout[(size_t)g * C3 + c] = pool[c];
}

// ---------------------------------------------------------------------------
// Dense linear + BN(eval) + ReLU: one wave per 16x16 output tile, K multiple
// of 4, M multiple of 16. grid = (N/16, M/16).
// ---------------------------------------------------------------------------
__global__ void __launch_bounds__(32)
linear_bn_relu_kernel(const float* __restrict__ X, LayerP L,
                      float* __restrict__ Y, int K, int N) {
  const int nt   = blockIdx.x;
  const int mt   = blockIdx.y;
  const int lane = threadIdx.x & 31;
  const int half = lane >> 4;
  const int l16  = lane & 15;
  v8f acc = vzero8();
  for (int kb = 0; kb < K; kb += 4) {
    const int k = kb + 2 * half;
    const float* xr = X + ((size_t)(mt * 16 + l16)) * K + k;
    v2f a, w;
    a.x = xr[0]; a.y = xr[1];
    w.x = L.W[(size_t)k       * N + nt * 16 + l16];
    w.y = L.W[(size_t)(k + 1) * N + nt * 16 + l16];
    acc = wmma_f32(a, w, acc);
  }
  const int n = nt * 16 + l16;
  const float sc = L.gm[n] * rsqrtf(L.var[n] + 1e-5f);
  const float sh = (L.b[n] - L.mu[n]) * sc + L.bt[n];
#pragma unroll
  for (int j = 0; j < 8; ++j)
    Y[(size_t)(mt * 16 + j + 8 * half) * N + n] = fmaxf(acc[j] * sc + sh, 0.f);
}

// ---------------------------------------------------------------------------
// Heads: q (normalized), t, sigmoid(conf). One wave per batch row.
// out layout: qt[64][7] flat (448 floats) followed by conf[64].
// ---------------------------------------------------------------------------
__global__ void __launch_bounds__(32)
heads_kernel(const float* __restrict__ z,
             const float* __restrict__ qW, const float* __restrict__ qb,
             const float* __restrict__ tW, const float* __restrict__ tb,
             const float* __restrict__ cW, const float* __restrict__ cb,
             float* __restrict__ out) {
  const int b    = blockIdx.x;
  const int lane = threadIdx.x;
  float acc[8];
#pragma unroll
  for (int o = 0; o < 8; ++o) acc[o] = 0.f;
  for (int k = lane; k < 512; k += 32) {
    const float zv = z[b * 512 + k];
#pragma unroll
    for (int o = 0; o < 4; ++o) acc[o]     += zv * qW[k * 4 + o];
#pragma unroll
    for (int o = 0; o < 3; ++o) acc[4 + o] += zv * tW[k * 3 + o];
    acc[7] += zv * cW[k];
  }
#pragma unroll
  for (int o = 0; o < 8; ++o) {
#pragma unroll
    for (int m = 16; m > 0; m >>= 1) acc[o] += __shfl_xor(acc[o], m);
  }
  if (lane == 0) {
    const float q0 = acc[0] + qb[0], q1 = acc[1] + qb[1];
    const float q2 = acc[2] + qb[2], q3 = acc[3] + qb[3];
    float nrm = sqrtf(q0 * q0 + q1 * q1 + q2 * q2 + q3 * q3);
    nrm = fmaxf(nrm, 1e-12f);
    out[b * 7 + 0] = q0 / nrm;
    out[b * 7 + 1] = q1 / nrm;
    out[b * 7 + 2] = q2 / nrm;
    out[b * 7 + 3] = q3 / nrm;
    out[b * 7 + 4] = acc[4] + tb[0];
    out[b * 7 + 5] = acc[5] + tb[1];
    out[b * 7 + 6] = acc[6] + tb[2];
    const float c = acc[7] + cb[0];
    out[448 + b] = 1.f / (1.f + expf(-c));
  }
}

// ---------------------------------------------------------------------------
extern "C" void kernel_launch(void* const* d_in, const int* in_sizes, int n_in,
                              void* d_out, int out_size, void* d_ws,
                              size_t ws_size, hipStream_t stream) {
  (void)in_sizes; (void)n_in; (void)out_size; (void)ws_size;
  const float* pc = (const float*)d_in[0];

  LayerP sa1[3], sa2[3], sa3[3], fc[2];
  int p = 1;
  auto take = [&](LayerP* L, int cnt) {
    for (int i = 0; i < cnt; ++i) {
      L[i].W   = (const float*)d_in[p++];
      L[i].b   = (const float*)d_in[p++];
      L[i].gm  = (const float*)d_in[p++];
      L[i].bt  = (const float*)d_in[p++];
      L[i].mu  = (const float*)d_in[p++];
      L[i].var = (const float*)d_in[p++];
    }
  };
  take(sa1, 3); take(sa2, 3); take(sa3, 3); take(fc, 2);
  const float* qW = (const float*)d_in[p++];
  const float* qb = (const float*)d_in[p++];
  const float* tW = (const float*)d_in[p++];
  const float* tb = (const float*)d_in[p++];
  const float* cW = (const float*)d_in[p++];
  const float* cb = (const float*)d_in[p++];

  char* ws = (char*)d_ws;
  size_t off = 0;
  auto alloc = [&](size_t bytes) -> void* {
    void* r = ws + off;
    off = (off + bytes + 255) & ~(size_t)255;
    return r;
  };
  int*   cidx1 = (int*)  alloc((size_t)64 * 128 * 4);
  float* xyz1  = (float*)alloc((size_t)64 * 128 * 3 * 4);
  int*   gidx1 = (int*)  alloc((size_t)64 * 128 * 64 * 4);
  float* f1    = (float*)alloc((size_t)64 * 128 * 128 * 4);
  int*   cidx2 = (int*)  alloc((size_t)64 * 32 * 4);
  float* xyz2  = (float*)alloc((size_t)64 * 32 * 3 * 4);
  int*   gidx2 = (int*)  alloc((size_t)64 * 32 * 128 * 4);
  float* f2    = (float*)alloc((size_t)64 * 32 * 256 * 4);
  float* z0    = (float*)alloc((size_t)64 * 512 * 4);
  float* z1    = (float*)alloc((size_t)64 * 1024 * 4);
  float* z2    = (float*)alloc((size_t)64 * 512 * 4);

  // ---- SA1 ----
  fps_kernel<4096, 128, 256><<<64, 256, 0, stream>>>(pc, cidx1);
  gather_xyz_kernel<<<(8192 + 255) / 256, 256, 0, stream>>>(pc, cidx1, xyz1,
                                                            4096, 128, 8192);
  ball_query_kernel<<<(8192 + 255) / 256, 256, 0, stream>>>(
      pc, xyz1, gidx1, 4096, 128, 64, 0.02f * 0.02f, 8192);
  sa_stage_kernel<1, 128, 64, 4096, 6, 64, 64, 128, 3>
      <<<8192, 32, 0, stream>>>(pc, xyz1, gidx1, pc, sa1[0], sa1[1], sa1[2], f1);

  // ---- SA2 ----
  fps_kernel<128, 32, 128><<<64, 128, 0, stream>>>(xyz1, cidx2);
  gather_xyz_kernel<<<(2048 + 255) / 256, 256, 0, stream>>>(xyz1, cidx2, xyz2,
                                                            128, 32, 2048);
  ball_query_kernel<<<(2048 + 255) / 256, 256, 0, stream>>>(
      xyz1, xyz2, gidx2, 128, 32, 128, 0.04f * 0.04f, 2048);
  sa_stage_kernel<2, 32, 128, 128, 131, 128, 128, 256, 128>
      <<<2048, 32, 0, stream>>>(xyz1, xyz2, gidx2, f1, sa2[0], sa2[1], sa2[2], f2);

  // ---- SA3 (global group of 32 points, input = concat(xyz2, f2)) ----
  sa_stage_kernel<3, 1, 32, 32, 259, 256, 256, 512, 256>
      <<<64, 32, 0, stream>>>(nullptr, xyz2, nullptr, f2, sa3[0], sa3[1],
                              sa3[2], z0);

  // ---- FC stack ----
  linear_bn_relu_kernel<<<dim3(1024 / 16, 64 / 16), 32, 0, stream>>>(
      z0, fc[0], z1, 512, 1024);
  linear_bn_relu_kernel<<<dim3(512 / 16, 64 / 16), 32, 0, stream>>>(
      z1, fc[1], z2, 1024, 512);

  // ---- Heads ----
  heads_kernel<<<64, 32, 0, stream>>>(z2, qW, qb, tW, tb, cW, cb,
                                      (float*)d_out);
}